// InteractionBlock_81724637708437
// MI455X (gfx1250) — compile-verified
//
#include <hip/hip_runtime.h>
#include <math.h>

// ---------------------------------------------------------------------------
// Equivariant interaction block for MI455X (gfx1250, wave32, WMMA).
// All channel-mixing matmuls are mapped to chains of V_WMMA_F32_16X16X4_F32
// over 16-row tiles (16 nodes or 16 edges). fp32 end-to-end (matches ref).
//
// Assumed d_in order (setup_inputs insertion order; params flattened with
// sorted dict keys, jax tree order):
//   0 nodes[8192,240] 1 pos[8192,3] 2 batch 3 edge_index[2,131072]
//   4 conv.Wr1[8,64] 5 conv.Wr2[64,480] 6 conv.br1[64] 7 conv.br2[480]
//   8..22 conv.proj['000','011','022','101','110','111','112','121','122',
//                   '202','211','212','220','221','222']
//   23 gate.Wg1[64,32] 24 gate.Wg2[64,16] 25 gate.bg1[32] 26 gate.bg2[16]
//   27 lin1.W0[64,64] 28 lin1.W1[32,32] 29 lin1.W2[16,16] 30 lin1.b0[64]
//   31 lin2.W0 32 lin2.W1 33 lin2.W2 34 lin2.b0
// ---------------------------------------------------------------------------

#define NN 8192
#define NE 131072
#define C0 64
#define C1 32
#define C2 16
#define FDIM 240          // 64 + 96 + 80
#define WTOT 480

typedef float v2f __attribute__((ext_vector_type(2)));
typedef float v8f __attribute__((ext_vector_type(8)));

__device__ __forceinline__ v8f wmma4(v2f a, v2f b, v8f c) {
  // D = A(16x4) * B(4x16) + C, fp32 matrix pipe
  return __builtin_amdgcn_wmma_f32_16x16x4_f32(false, a, false, b, (short)0, c,
                                               false, false);
}

__device__ __forceinline__ int accM(int i, int lane) {
  // C/D layout: VGPR i, lanes 0-15 -> M=i ; lanes 16-31 -> M=i+8
  return i + ((lane >> 4) << 3);
}

// Chained WMMA: acc[t] += A @ B[:, t*16:(t+1)*16], A generated per-element.
// A layout (16x4 f32): lane (h*16+m) holds A[m][kbase+2h .. +1] in 2 VGPRs.
// B layout (4x16 f32): lane (h*16+n) holds B[kbase+2h .. +1][n].
template <int CIN, int NT, typename FA>
__device__ __forceinline__ void mm_f32(FA getA, const float* __restrict__ B,
                                       int ldb, v8f* acc, int lane) {
  const int m = lane & 15;
  const int kh = (lane >> 4) * 2;
  for (int ks = 0; ks < CIN; ks += 4) {
    v2f a;
    a.x = getA(ks + kh);
    a.y = getA(ks + kh + 1);
#pragma unroll
    for (int t = 0; t < NT; ++t) {
      v2f b;
      b.x = B[(ks + kh) * ldb + t * 16 + m];
      b.y = B[(ks + kh + 1) * ldb + t * 16 + m];
      acc[t] = wmma4(a, b, acc[t]);
    }
  }
}

__device__ __forceinline__ float siluf(float z) { return z / (1.f + __expf(-z)); }
__device__ __forceinline__ float sigmf(float z) { return 1.f / (1.f + __expf(-z)); }

// ============================ kernel 1: lin1 ================================
__global__ __launch_bounds__(32) void k_lin1(
    const float* __restrict__ nodes, const float* __restrict__ W0,
    const float* __restrict__ W1, const float* __restrict__ W2,
    const float* __restrict__ b0, float* __restrict__ s1,
    float* __restrict__ v1, float* __restrict__ T1, float* __restrict__ cs,
    float* __restrict__ cv, float* __restrict__ cT) {
  const int lane = threadIdx.x;
  const int m = lane & 15;
  const int n0 = blockIdx.x * 16;
  const float* row = nodes + (size_t)(n0 + m) * FDIM;

  // zero convolution accumulators for this node tile
  for (int i = lane; i < 16 * 64; i += 32) cs[(size_t)n0 * 64 + i] = 0.f;
  for (int i = lane; i < 16 * 96; i += 32) cv[(size_t)n0 * 96 + i] = 0.f;
  for (int i = lane; i < 16 * 144; i += 32) cT[(size_t)n0 * 144 + i] = 0.f;

  // s1 = s @ W0 + b0   (16x64 @ 64x64)
  {
    v8f a[4] = {};
    mm_f32<64, 4>([&](int k) { return row[k]; }, W0, 64, a, lane);
#pragma unroll
    for (int t = 0; t < 4; ++t)
#pragma unroll
      for (int i = 0; i < 8; ++i) {
        int M = accM(i, lane), N = m + t * 16;
        s1[(size_t)(n0 + M) * 64 + N] = a[t][i] + b0[N];
      }
  }
  // v1[:,d,x] = sum_c v[:,c,x] W1[c,d]
  for (int x = 0; x < 3; ++x) {
    v8f a[2] = {};
    mm_f32<32, 2>([&](int k) { return row[64 + k * 3 + x]; }, W1, 32, a, lane);
#pragma unroll
    for (int t = 0; t < 2; ++t)
#pragma unroll
      for (int i = 0; i < 8; ++i) {
        int M = accM(i, lane), d = m + t * 16;
        v1[(size_t)(n0 + M) * 96 + d * 3 + x] = a[t][i];
      }
  }
  // t1 per spherical component, then assemble 3x3 traceless-symmetric T
  {
    v8f ta[5];
#pragma unroll
    for (int a5 = 0; a5 < 5; ++a5) {
      v8f acc = {};
      mm_f32<16, 1>([&](int k) { return row[160 + k * 5 + a5]; }, W2, 16, &acc,
                    lane);
      ta[a5] = acc;
    }
    const float s2c = 0.70710678118654752f;   // 1/sqrt(2)
    const float s6c = 0.40824829046386302f;   // 1/sqrt(6)
#pragma unroll
    for (int i = 0; i < 8; ++i) {
      int M = accM(i, lane), c = m;
      float t0 = ta[0][i], t1 = ta[1][i], t2 = ta[2][i], t3 = ta[3][i],
            t4 = ta[4][i];
      float* T = T1 + (size_t)(n0 + M) * 144 + c * 9;
      T[0] = s2c * t3 - s6c * t4;          // 00
      T[4] = -s2c * t3 - s6c * t4;         // 11
      T[8] = 2.f * s6c * t4;               // 22
      T[1] = s2c * t0; T[3] = s2c * t0;    // 01,10
      T[5] = s2c * t1; T[7] = s2c * t1;    // 12,21
      T[2] = s2c * t2; T[6] = s2c * t2;    // 02,20
    }
  }
}

// ============================ kernel 2: conv ================================
struct ProjPtrs { const float* p[15]; };
// proj order: 0:000 1:011 2:022 3:101 4:110 5:111 6:112 7:121 8:122
//             9:202 10:211 11:212 12:220 13:221 14:222
// wall slice offsets (PATH_ORDER cumulative cin):
//  000:0 011:64 022:128 101:192 110:224 111:256 112:288 121:320 122:352
//  202:384 211:400 212:416 220:432 221:448 222:464

__global__ __launch_bounds__(32) void k_conv(
    const int* __restrict__ eidx, const float* __restrict__ pos,
    const float* __restrict__ s1, const float* __restrict__ v1g,
    const float* __restrict__ T1, const float* __restrict__ Wr1,
    const float* __restrict__ br1, const float* __restrict__ Wr2,
    const float* __restrict__ br2, ProjPtrs PR, float* __restrict__ cs,
    float* __restrict__ cv, float* __restrict__ cT) {
  __shared__ float su[16 * 3];
  __shared__ float sY2[16 * 9];
  __shared__ int ssrc[16];
  __shared__ int sdst[16];
  __shared__ float sh[16 * 64];
  __shared__ float sw[16 * WTOT];
  __shared__ float sTY[16 * 144];

  const int lane = threadIdx.x;
  const int m = lane & 15;
  const int e = blockIdx.x * 16 + m;

  // ---- per-edge geometry (both half-waves compute their row's edge) ----
  const int nsrc = eidx[e];
  const int ndst = eidx[NE + e];
  float rx = pos[ndst * 3 + 0] - pos[nsrc * 3 + 0];
  float ry = pos[ndst * 3 + 1] - pos[nsrc * 3 + 1];
  float rz = pos[ndst * 3 + 2] - pos[nsrc * 3 + 2];
  float d = sqrtf(rx * rx + ry * ry + rz * rz);
  float inv = 1.f / fmaxf(d, 1e-6f);
  float u0 = rx * inv, u1 = ry * inv, u2 = rz * inv;
  if (lane < 16) {
    ssrc[m] = nsrc;
    sdst[m] = ndst;
    su[m * 3 + 0] = u0; su[m * 3 + 1] = u1; su[m * 3 + 2] = u2;
    float uu[3] = {u0, u1, u2};
#pragma unroll
    for (int i = 0; i < 3; ++i)
#pragma unroll
      for (int j = 0; j < 3; ++j)
        sY2[m * 9 + i * 3 + j] = uu[i] * uu[j] - (i == j ? (1.f / 3.f) : 0.f);
  }

  // ---- radial MLP layer 1: h = silu(rbf @ Wr1 + br1) ----
  {
    v8f h[4] = {};
    mm_f32<8, 4>(
        [&](int k) {
          float t = (d - (float)k * (3.0f / 7.0f)) * (8.0f / 3.0f);
          return __expf(-t * t);
        },
        Wr1, 64, h, lane);
#pragma unroll
    for (int t = 0; t < 4; ++t)
#pragma unroll
      for (int i = 0; i < 8; ++i) {
        int M = accM(i, lane), N = m + t * 16;
        sh[M * 64 + N] = siluf(h[t][i] + br1[N]);
      }
  }
  __syncthreads();

  // ---- radial MLP layer 2: wall = h @ Wr2 + br2 (30 output tiles) ----
  for (int t = 0; t < 30; ++t) {
    v8f acc = {};
    mm_f32<64, 1>([&](int k) { return sh[m * 64 + k]; }, Wr2 + t * 16, WTOT,
                  &acc, lane);
#pragma unroll
    for (int i = 0; i < 8; ++i) {
      int M = accM(i, lane), N = m + t * 16;
      sw[M * WTOT + N] = acc[i] + br2[N];
    }
  }
  __syncthreads();

  // ---- stage TY = T(src) @ Y2 per edge/channel ----
  for (int idx = lane; idx < 16 * 16 * 9; idx += 32) {
    int em = idx / 144, r = idx - em * 144;
    int c = r / 9, ij = r - c * 9, i = ij / 3, j = ij - i * 3;
    const float* Tp = T1 + (size_t)ssrc[em] * 144 + c * 9;
    sTY[idx] = Tp[i * 3 + 0] * sY2[em * 9 + 0 + j] +
               Tp[i * 3 + 1] * sY2[em * 9 + 3 + j] +
               Tp[i * 3 + 2] * sY2[em * 9 + 6 + j];
  }
  __syncthreads();

  const float* wrow = &sw[m * WTOT];
  const float* srow = s1 + (size_t)nsrc * 64;
  const float* vrow = v1g + (size_t)nsrc * 96;
  const float* Trow = T1 + (size_t)nsrc * 144;
  const float* TYrow = &sTY[m * 144];
  const float* urow = &su[m * 3];
  const float* yrow = &sY2[m * 9];

  // ================= m0 : scalar messages (paths 000,110,220) =============
  {
    v8f a0[4] = {};
    mm_f32<64, 4>([&](int k) { return wrow[0 + k] * srow[k]; }, PR.p[0], 64,
                  a0, lane);
    mm_f32<32, 4>(
        [&](int k) {
          const float* vp = vrow + k * 3;
          return wrow[224 + k] *
                 (vp[0] * urow[0] + vp[1] * urow[1] + vp[2] * urow[2]);
        },
        PR.p[4], 64, a0, lane);
    mm_f32<16, 4>(
        [&](int k) {
          const float* Tp = Trow + k * 9;
          float s = 0.f;
#pragma unroll
          for (int q = 0; q < 9; ++q) s += Tp[q] * yrow[q];
          return wrow[432 + k] * s;
        },
        PR.p[12], 64, a0, lane);
#pragma unroll
    for (int t = 0; t < 4; ++t)
#pragma unroll
      for (int i = 0; i < 8; ++i) {
        int M = accM(i, lane), N = m + t * 16;
        atomicAdd(&cs[(size_t)sdst[M] * 64 + N], 0.25f * a0[t][i]);
      }
  }

  // ================= m1 : vector messages ==================================
  {
    v8f r011[2] = {};
    mm_f32<64, 2>([&](int k) { return wrow[64 + k] * srow[k]; }, PR.p[1], 32,
                  r011, lane);
    for (int x = 0; x < 3; ++x) {
      const int x1 = (x + 1) % 3, x2 = (x + 2) % 3;
      v8f a1[2] = {};
      mm_f32<32, 2>([&](int k) { return wrow[192 + k] * vrow[k * 3 + x]; },
                    PR.p[3], 32, a1, lane);  // 101
      mm_f32<32, 2>(
          [&](int k) {
            const float* vp = vrow + k * 3;
            return wrow[256 + k] * (vp[x1] * urow[x2] - vp[x2] * urow[x1]);
          },
          PR.p[5], 32, a1, lane);  // 111 cross(v,u)
      mm_f32<32, 2>(
          [&](int k) {
            const float* vp = vrow + k * 3;
            return wrow[320 + k] * (yrow[x * 3 + 0] * vp[0] +
                                    yrow[x * 3 + 1] * vp[1] +
                                    yrow[x * 3 + 2] * vp[2]);
          },
          PR.p[7], 32, a1, lane);  // 121 Y2.v
      mm_f32<16, 2>(
          [&](int k) {
            const float* Tp = Trow + k * 9;
            return wrow[400 + k] * (Tp[x * 3 + 0] * urow[0] +
                                    Tp[x * 3 + 1] * urow[1] +
                                    Tp[x * 3 + 2] * urow[2]);
          },
          PR.p[10], 32, a1, lane);  // 211 T.u
      mm_f32<16, 2>(
          [&](int k) {
            const float* TYp = TYrow + k * 9;
            return wrow[448 + k] * (TYp[x1 * 3 + x2] - TYp[x2 * 3 + x1]);
          },
          PR.p[13], 32, a1, lane);  // 221 dual(T@Y2)
#pragma unroll
      for (int t = 0; t < 2; ++t)
#pragma unroll
        for (int i = 0; i < 8; ++i) {
          int M = accM(i, lane), c = m + t * 16;
          float val = a1[t][i] + r011[t][i] * su[M * 3 + x];  // + 011 term
          atomicAdd(&cv[(size_t)sdst[M] * 96 + c * 3 + x], 0.25f * val);
        }
    }
  }

  // ================= m2 : rank-2 messages ==================================
  {
    v8f r022 = {};
    mm_f32<64, 1>([&](int k) { return wrow[128 + k] * srow[k]; }, PR.p[2], 16,
                  &r022, lane);
    for (int ij = 0; ij < 9; ++ij) {
      const int i3 = ij / 3, j3 = ij % 3;
      const int i1 = (i3 + 1) % 3, i2 = (i3 + 2) % 3;
      const int j1 = (j3 + 1) % 3, j2 = (j3 + 2) % 3;
      v8f a2 = {};
      mm_f32<32, 1>(
          [&](int k) {  // 112 sym_traceless(v (x) u)
            const float* vp = vrow + k * 3;
            float s = 0.5f * (vp[i3] * urow[j3] + vp[j3] * urow[i3]);
            if (i3 == j3)
              s -= (vp[0] * urow[0] + vp[1] * urow[1] + vp[2] * urow[2]) *
                   (1.f / 3.f);
            return wrow[288 + k] * s;
          },
          PR.p[6], 16, &a2, lane);
      mm_f32<32, 1>(
          [&](int k) {  // 122 sym(cross_cols(v, Y2))
            const float* vp = vrow + k * 3;
            float Mij = vp[i1] * yrow[i2 * 3 + j3] - vp[i2] * yrow[i1 * 3 + j3];
            float Mji = vp[j1] * yrow[j2 * 3 + i3] - vp[j2] * yrow[j1 * 3 + i3];
            return wrow[352 + k] * 0.5f * (Mij + Mji);
          },
          PR.p[8], 16, &a2, lane);
      mm_f32<16, 1>([&](int k) { return wrow[384 + k] * Trow[k * 9 + ij]; },
                    PR.p[9], 16, &a2, lane);  // 202
      mm_f32<16, 1>(
          [&](int k) {  // 212 sym(cross_cols(u, T))
            const float* Tp = Trow + k * 9;
            float Mij = urow[i1] * Tp[i2 * 3 + j3] - urow[i2] * Tp[i1 * 3 + j3];
            float Mji = urow[j1] * Tp[j2 * 3 + i3] - urow[j2] * Tp[j1 * 3 + i3];
            return wrow[416 + k] * 0.5f * (Mij + Mji);
          },
          PR.p[11], 16, &a2, lane);
      mm_f32<16, 1>(
          [&](int k) {  // 222 sym_traceless(T@Y2)
            const float* TYp = TYrow + k * 9;
            float s = 0.5f * (TYp[ij] + TYp[j3 * 3 + i3]);
            if (i3 == j3) s -= (TYp[0] + TYp[4] + TYp[8]) * (1.f / 3.f);
            return wrow[464 + k] * s;
          },
          PR.p[14], 16, &a2, lane);
#pragma unroll
      for (int i = 0; i < 8; ++i) {
        int M = accM(i, lane), c = m;
        float val = a2[i] + r022[i] * sY2[M * 9 + ij];  // + 022 term
        atomicAdd(&cT[(size_t)sdst[M] * 144 + c * 9 + ij], 0.25f * val);
      }
    }
  }
}

// ======================= kernel 3: lin2 + gate + merge ======================
__global__ __launch_bounds__(32) void k_lin2_gate(
    const float* __restrict__ nodes, const float* __restrict__ cs,
    const float* __restrict__ cv, const float* __restrict__ cT,
    const float* __restrict__ W0, const float* __restrict__ W1,
    const float* __restrict__ W2, const float* __restrict__ b0,
    const float* __restrict__ Wg1, const float* __restrict__ bg1,
    const float* __restrict__ Wg2, const float* __restrict__ bg2,
    float* __restrict__ out) {
  __shared__ float ssg[16 * 64];
  const int lane = threadIdx.x;
  const int m = lane & 15;
  const int n0 = blockIdx.x * 16;
  const float s2c = 0.70710678118654752f, s6c = 0.40824829046386302f;

  // sg = s_orig + (cs @ W0 + b0); emit silu(sg) and stage sg in LDS
  {
    v8f a[4] = {};
    mm_f32<64, 4>([&](int k) { return cs[(size_t)(n0 + m) * 64 + k]; }, W0, 64,
                  a, lane);
#pragma unroll
    for (int t = 0; t < 4; ++t)
#pragma unroll
      for (int i = 0; i < 8; ++i) {
        int M = accM(i, lane), N = m + t * 16;
        float z = a[t][i] + b0[N] + nodes[(size_t)(n0 + M) * FDIM + N];
        ssg[M * 64 + N] = z;
        out[(size_t)(n0 + M) * FDIM + N] = siluf(z);
      }
  }
  __syncthreads();

  // gates g1 = sigmoid(sg@Wg1+bg1), g2 = sigmoid(sg@Wg2+bg2)
  v8f g1[2] = {};
  mm_f32<64, 2>([&](int k) { return ssg[m * 64 + k]; }, Wg1, 32, g1, lane);
#pragma unroll
  for (int t = 0; t < 2; ++t)
#pragma unroll
    for (int i = 0; i < 8; ++i) g1[t][i] = sigmf(g1[t][i] + bg1[m + t * 16]);
  v8f g2 = {};
  mm_f32<64, 1>([&](int k) { return ssg[m * 64 + k]; }, Wg2, 16, &g2, lane);
#pragma unroll
  for (int i = 0; i < 8; ++i) g2[i] = sigmf(g2[i] + bg2[m]);

  // vectors: out_v = (v_orig + cv @ W1) * g1
  for (int x = 0; x < 3; ++x) {
    v8f a[2] = {};
    mm_f32<32, 2>([&](int k) { return cv[(size_t)(n0 + m) * 96 + k * 3 + x]; },
                  W1, 32, a, lane);
#pragma unroll
    for (int t = 0; t < 2; ++t)
#pragma unroll
      for (int i = 0; i < 8; ++i) {
        int M = accM(i, lane), c = m + t * 16;
        size_t o = (size_t)(n0 + M) * FDIM + 64 + c * 3 + x;
        out[o] = (a[t][i] + nodes[o]) * g1[t][i];
      }
  }

  // tensors: ct5 = mat_to_t5(cT) on the fly; out_t = (t_orig + ct5@W2) * g2
  for (int a5 = 0; a5 < 5; ++a5) {
    v8f a = {};
    mm_f32<16, 1>(
        [&](int k) {
          const float* Tp = cT + (size_t)(n0 + m) * 144 + k * 9;
          switch (a5) {
            case 0: return s2c * (Tp[1] + Tp[3]);
            case 1: return s2c * (Tp[5] + Tp[7]);
            case 2: return s2c * (Tp[2] + Tp[6]);
            case 3: return s2c * (Tp[0] - Tp[4]);
            default: return s6c * (2.f * Tp[8] - Tp[0] - Tp[4]);
          }
        },
        W2, 16, &a, lane);
#pragma unroll
    for (int i = 0; i < 8; ++i) {
      int M = accM(i, lane), c = m;
      size_t o = (size_t)(n0 + M) * FDIM + 160 + c * 5 + a5;
      out[o] = (a[i] + nodes[o]) * g2[i];
    }
  }
}

// ================================ launcher ==================================
extern "C" void kernel_launch(void* const* d_in, const int* in_sizes, int n_in,
                              void* d_out, int out_size, void* d_ws,
                              size_t ws_size, hipStream_t stream) {
  if (n_in < 35 || !d_out || !d_ws) return;
  const float* nodes = (const float*)d_in[0];
  const float* pos = (const float*)d_in[1];
  const int* eidx = (const int*)d_in[3];
  const float* Wr1 = (const float*)d_in[4];
  const float* Wr2 = (const float*)d_in[5];
  const float* br1 = (const float*)d_in[6];
  const float* br2 = (const float*)d_in[7];
  ProjPtrs PR;
  for (int i = 0; i < 15; ++i) PR.p[i] = (const float*)d_in[8 + i];
  const float* Wg1 = (const float*)d_in[23];
  const float* Wg2 = (const float*)d_in[24];
  const float* bg1 = (const float*)d_in[25];
  const float* bg2 = (const float*)d_in[26];
  const float* l1W0 = (const float*)d_in[27];
  const float* l1W1 = (const float*)d_in[28];
  const float* l1W2 = (const float*)d_in[29];
  const float* l1b0 = (const float*)d_in[30];
  const float* l2W0 = (const float*)d_in[31];
  const float* l2W1 = (const float*)d_in[32];
  const float* l2W2 = (const float*)d_in[33];
  const float* l2b0 = (const float*)d_in[34];

  // workspace partition (floats)
  float* ws = (float*)d_ws;
  float* s1 = ws;                        // 8192*64
  float* v1 = s1 + (size_t)NN * 64;      // 8192*96
  float* T1 = v1 + (size_t)NN * 96;      // 8192*144
  float* cs = T1 + (size_t)NN * 144;     // 8192*64
  float* cv = cs + (size_t)NN * 64;      // 8192*96
  float* cT = cv + (size_t)NN * 96;      // 8192*144
  (void)ws_size; (void)in_sizes; (void)out_size;

  k_lin1<<<dim3(NN / 16), dim3(32), 0, stream>>>(nodes, l1W0, l1W1, l1W2, l1b0,
                                                 s1, v1, T1, cs, cv, cT);
  k_conv<<<dim3(NE / 16), dim3(32), 0, stream>>>(eidx, pos, s1, v1, T1, Wr1,
                                                 br1, Wr2, br2, PR, cs, cv, cT);
  k_lin2_gate<<<dim3(NN / 16), dim3(32), 0, stream>>>(
      nodes, cs, cv, cT, l2W0, l2W1, l2W2, l2b0, Wg1, bg1, Wg2, bg2,
      (float*)d_out);
}